// PtModule_73667279061056
// MI455X (gfx1250) — compile-verified
//
#include <hip/hip_runtime.h>
#include <hip/hip_bf16.h>

// Memory-bound elementwise map:
//   out = (x + 2) + (x * 3) - (x - 1) * (x * 0.5)
// 512 MB in + 512 MB out, touched once -> stream through HBM with
// 128-bit loads/stores and non-temporal (NT) cache-policy hints so we
// don't pollute the 192 MB L2 with dead data.

typedef float v4f __attribute__((ext_vector_type(4)));

__device__ __forceinline__ float fused_map(float x) {
    return (x + 2.0f) + (x * 3.0f) - (x - 1.0f) * (x * 0.5f);
}

__global__ void __launch_bounds__(256)
elementwise_vec4_kernel(const float* __restrict__ x,
                        float* __restrict__ out,
                        long long n4) {
    long long i = (long long)blockIdx.x * blockDim.x + threadIdx.x;
    if (i < n4) {
        const v4f* __restrict__ xv = (const v4f*)x;
        v4f*       __restrict__ ov = (v4f*)out;

        // global_load_b128 with NT temporal hint (streaming, read-once)
        v4f v = __builtin_nontemporal_load(&xv[i]);

        v4f r;
        r.x = fused_map(v.x);
        r.y = fused_map(v.y);
        r.z = fused_map(v.z);
        r.w = fused_map(v.w);

        // global_store_b128 with NT temporal hint (write-once)
        __builtin_nontemporal_store(r, &ov[i]);
    }
}

__global__ void __launch_bounds__(256)
elementwise_tail_kernel(const float* __restrict__ x,
                        float* __restrict__ out,
                        long long start,
                        long long n) {
    long long i = start + (long long)blockIdx.x * blockDim.x + threadIdx.x;
    if (i < n) {
        float v = __builtin_nontemporal_load(&x[i]);
        __builtin_nontemporal_store(fused_map(v), &out[i]);
    }
}

extern "C" void kernel_launch(void* const* d_in, const int* in_sizes, int n_in,
                              void* d_out, int out_size, void* d_ws, size_t ws_size,
                              hipStream_t stream) {
    (void)n_in; (void)d_ws; (void)ws_size;

    const float* x = (const float*)d_in[0];
    float* out = (float*)d_out;

    const long long n  = (long long)out_size;   // 8192*16384 = 134217728
    const long long n4 = n >> 2;                // float4 chunks
    const long long rem_start = n4 << 2;

    if (n4 > 0) {
        const int block = 256;
        const long long grid = (n4 + block - 1) / block;   // 131072 blocks
        elementwise_vec4_kernel<<<(unsigned)grid, block, 0, stream>>>(x, out, n4);
    }
    if (rem_start < n) {
        const int block = 256;
        const long long tail = n - rem_start;
        const long long grid = (tail + block - 1) / block;
        elementwise_tail_kernel<<<(unsigned)grid, block, 0, stream>>>(x, out, rem_start, n);
    }
}